// TtMixtralAttention_47304769798244
// MI455X (gfx1250) — compile-verified
//
#include <hip/hip_runtime.h>
#include <hip/hip_bf16.h>

typedef __attribute__((ext_vector_type(2))) float v2f;
typedef __attribute__((ext_vector_type(8))) float v8f;

#define NHEADS   32
#define NKV      8
#define HD       128
#define DIMSZ    4096
#define WIN      4096
#define BATCH    32
#define QKVCOLS  6144   /* (32 + 2*8) * 128 */

// ---------------------------------------------------------------------------
// GEMM: C[32 x N] = A[32 x K] * B[K x N]   (fp32, row-major), K == 4096.
// Block = 256 threads = 8 waves.  Each block owns a 32-column chunk; the 8
// waves split K into slices of K/8 and each computes a partial 32x32 tile
// (2 M-tiles x 2 N-tiles of V_WMMA_F32_16X16X4_F32).  Partials are reduced
// deterministically through LDS.  Weight bytes are read exactly once, and
// the launch exposes ~1.5K waves so the 100 MB weight stream can actually
// use HBM bandwidth (single-wave streaming tops out near ~10 GB/s).
// A-frag layout (ISA 7.12.2, 32-bit A 16x4): lanes 0-15 row M=lane K={0,1},
// lanes 16-31 row M=lane-16 K={2,3}.  B mirrored (transposed) layout.
// C/D: VGPR r -> M = r + (lane>=16 ? 8 : 0), N = lane & 15.
// ---------------------------------------------------------------------------
__global__ void gemm32_wmma(const float* __restrict__ A,
                            const float* __restrict__ B,
                            float* __restrict__ C, int K, int N) {
  __shared__ float part[8 * 32 * 32];  // 8 waves x (32 lanes x 32 vals) = 32 KB

  const int lane = threadIdx.x & 31;
  const int wave = threadIdx.x >> 5;       // 0..7 : K slice
  const int n0   = blockIdx.x * 32;

  const int r  = lane & 15;                // row-in-tile (A) / col-in-tile (B,C)
  const int kh = (lane >> 4) << 1;         // 0 or 2 : which K pair this lane holds

  const int kslice = K >> 3;
  const int kbeg   = wave * kslice;
  const int kend   = kbeg + kslice;

  v8f acc[2][2];
#pragma unroll
  for (int m = 0; m < 2; ++m)
#pragma unroll
    for (int n = 0; n < 2; ++n)
      acc[m][n] = (v8f){0.f, 0.f, 0.f, 0.f, 0.f, 0.f, 0.f, 0.f};

  const float* a0p = A + (size_t)r * K + kh;
  const float* a1p = A + (size_t)(r + 16) * K + kh;
  const float* bp  = B + (size_t)kh * N + n0 + r;

  for (int k0 = kbeg; k0 < kend; k0 += 4) {
    v2f a0 = {a0p[k0], a0p[k0 + 1]};
    v2f a1 = {a1p[k0], a1p[k0 + 1]};
    const float* brow = bp + (size_t)k0 * N;
    v2f b0 = {brow[0],  brow[N]};
    v2f b1 = {brow[16], brow[16 + N]};
    acc[0][0] = __builtin_amdgcn_wmma_f32_16x16x4_f32(
        false, a0, false, b0, (short)0, acc[0][0], false, false);
    acc[0][1] = __builtin_amdgcn_wmma_f32_16x16x4_f32(
        false, a0, false, b1, (short)0, acc[0][1], false, false);
    acc[1][0] = __builtin_amdgcn_wmma_f32_16x16x4_f32(
        false, a1, false, b0, (short)0, acc[1][0], false, false);
    acc[1][1] = __builtin_amdgcn_wmma_f32_16x16x4_f32(
        false, a1, false, b1, (short)0, acc[1][1], false, false);
  }

  // dump 32 partial floats per lane: j = m*16 + n*8 + rr
  float* my = part + (size_t)(wave * 32 + lane) * 32;
#pragma unroll
  for (int m = 0; m < 2; ++m)
#pragma unroll
    for (int n = 0; n < 2; ++n)
#pragma unroll
      for (int rr = 0; rr < 8; ++rr)
        my[m * 16 + n * 8 + rr] = acc[m][n][rr];
  __syncthreads();

  // deterministic 8-way reduce + store: thread t owns (L = t>>3, 4 j values)
  const int L  = threadIdx.x >> 3;
  const int j0 = (threadIdx.x & 7) * 4;
#pragma unroll
  for (int jj = 0; jj < 4; ++jj) {
    const int j = j0 + jj;
    float s = 0.f;
#pragma unroll
    for (int w = 0; w < 8; ++w) s += part[(size_t)(w * 32 + L) * 32 + j];
    const int m  = j >> 4;
    const int n  = (j >> 3) & 1;
    const int rr = j & 7;
    const int row = m * 16 + ((L >> 4) << 3) + rr;
    const int col = n0 + n * 16 + (L & 15);
    C[(size_t)row * N + col] = s;
  }
}

// ---------------------------------------------------------------------------
// Rotary: for each batch row, out[h][e] = sum_d qkv[h*128+d] * rot[d][e]
// q heads (h<32) additionally scaled by 1/sqrt(128).  Grid = 32, block = 128.
// ---------------------------------------------------------------------------
__global__ void rotary_kernel(const float* __restrict__ qkv,
                              const float* __restrict__ rot,
                              float* __restrict__ qrot,
                              float* __restrict__ krot) {
  __shared__ float xr[(NHEADS + NKV) * HD];  // 5120 floats = 20 KB
  const int b = blockIdx.x;
  const int e = threadIdx.x;  // 0..127

  for (int i = e; i < (NHEADS + NKV) * HD; i += 128)
    xr[i] = qkv[(size_t)b * QKVCOLS + i];
  __syncthreads();

  float acc[NHEADS + NKV];
#pragma unroll
  for (int h = 0; h < NHEADS + NKV; ++h) acc[h] = 0.f;

  for (int d = 0; d < HD; ++d) {
    const float rv = rot[d * HD + e];   // coalesced across lanes
#pragma unroll
    for (int h = 0; h < NHEADS + NKV; ++h)
      acc[h] = fmaf(xr[h * HD + d], rv, acc[h]);  // LDS broadcast
  }

  const float scale = 0.08838834764831845f;  // 128^-0.5
  for (int h = 0; h < NHEADS; ++h)
    qrot[(size_t)b * NHEADS * HD + h * HD + e] = acc[h] * scale;
  for (int h = 0; h < NKV; ++h)
    krot[(size_t)b * NKV * HD + h * HD + e] = acc[NHEADS + h];
}

// ---------------------------------------------------------------------------
// Attention: one block per (kv_head, batch).  4 grouped q-heads.
// Dynamic LDS layout (floats):
//   scores[4*4096] | qs[512] | red[256] | sums[1024] | inv[4] + pad
// ---------------------------------------------------------------------------
__global__ void attn_kernel(const float* __restrict__ qrot,
                            const float* __restrict__ krot,
                            const float* __restrict__ qkv,
                            const float* __restrict__ cache_k,
                            const float* __restrict__ cache_v,
                            const int* __restrict__ start_pos_p,
                            float* __restrict__ attn_out) {
  extern __shared__ float smem[];
  float* scores = smem;              // 4 * 4096
  float* qs     = scores + 4 * WIN;  // 4 * 128
  float* red    = qs + 512;          // 4 * 64
  float* sums   = red + 256;         // 4 * 256
  float* inv    = sums + 1024;       // 4

  const int tid = threadIdx.x;       // 256 threads
  const int b   = blockIdx.x & (BATCH - 1);
  const int kv  = blockIdx.x >> 5;

  const int sp  = start_pos_p[0];
  const int cur = sp & (WIN - 1);
  const int L   = (sp + 1 < WIN) ? (sp + 1) : WIN;

  // load the 4 grouped q heads (already rot+scaled)
  for (int i = tid; i < 4 * HD; i += 256)
    qs[i] = qrot[(size_t)b * NHEADS * HD + (size_t)kv * 4 * HD + i];
  __syncthreads();

  const float* kbase = cache_k + ((size_t)(kv * BATCH + b)) * WIN * HD;
  const float* knew  = krot + (size_t)b * NKV * HD + (size_t)kv * HD;

  // ---- score pass: stream K exactly once -------------------------------
  for (int p = tid; p < WIN; p += 256) {
    const int pn = p + 256;
    if (pn < L) __builtin_prefetch(kbase + (size_t)pn * HD, 0, 0);
    if (p >= L) {
#pragma unroll
      for (int g = 0; g < 4; ++g) scores[g * WIN + p] = -3.4e38f;
      continue;
    }
    const float4* kr = (p == cur) ? (const float4*)knew
                                  : (const float4*)(kbase + (size_t)p * HD);
    float a0 = 0.f, a1 = 0.f, a2 = 0.f, a3 = 0.f;
    const float4* q0 = (const float4*)(qs);
    const float4* q1 = (const float4*)(qs + HD);
    const float4* q2 = (const float4*)(qs + 2 * HD);
    const float4* q3 = (const float4*)(qs + 3 * HD);
#pragma unroll 4
    for (int i = 0; i < HD / 4; ++i) {
      const float4 kk = kr[i];
      const float4 w0 = q0[i], w1 = q1[i], w2 = q2[i], w3 = q3[i];
      a0 += kk.x * w0.x + kk.y * w0.y + kk.z * w0.z + kk.w * w0.w;
      a1 += kk.x * w1.x + kk.y * w1.y + kk.z * w1.z + kk.w * w1.w;
      a2 += kk.x * w2.x + kk.y * w2.y + kk.z * w2.z + kk.w * w2.w;
      a3 += kk.x * w3.x + kk.y * w3.y + kk.z * w3.z + kk.w * w3.w;
    }
    scores[0 * WIN + p] = a0;
    scores[1 * WIN + p] = a1;
    scores[2 * WIN + p] = a2;
    scores[3 * WIN + p] = a3;
  }
  __syncthreads();

  // ---- softmax: block-wide max per head --------------------------------
  {
    const int g = tid >> 6, j = tid & 63;
    float m = -3.4e38f;
    for (int p = j; p < WIN; p += 64) m = fmaxf(m, scores[g * WIN + p]);
    red[g * 64 + j] = m;
    __syncthreads();
    for (int off = 32; off >= 1; off >>= 1) {
      if (j < off) red[g * 64 + j] = fmaxf(red[g * 64 + j], red[g * 64 + j + off]);
      __syncthreads();
    }
  }
  const float m0 = red[0], m1 = red[64], m2 = red[128], m3 = red[192];

  // ---- exp + partial sums ----------------------------------------------
  float ls0 = 0.f, ls1 = 0.f, ls2 = 0.f, ls3 = 0.f;
  for (int p = tid; p < WIN; p += 256) {
    float e0 = __expf(scores[0 * WIN + p] - m0);
    float e1 = __expf(scores[1 * WIN + p] - m1);
    float e2 = __expf(scores[2 * WIN + p] - m2);
    float e3 = __expf(scores[3 * WIN + p] - m3);
    scores[0 * WIN + p] = e0; ls0 += e0;
    scores[1 * WIN + p] = e1; ls1 += e1;
    scores[2 * WIN + p] = e2; ls2 += e2;
    scores[3 * WIN + p] = e3; ls3 += e3;
  }
  sums[0 * 256 + tid] = ls0;
  sums[1 * 256 + tid] = ls1;
  sums[2 * 256 + tid] = ls2;
  sums[3 * 256 + tid] = ls3;
  __syncthreads();
  for (int off = 128; off >= 1; off >>= 1) {
    if (tid < off) {
#pragma unroll
      for (int g = 0; g < 4; ++g)
        sums[g * 256 + tid] += sums[g * 256 + tid + off];
    }
    __syncthreads();
  }
  if (tid < 4) inv[tid] = 1.f / sums[tid * 256];
  __syncthreads();

  // ---- output pass: stream V exactly once ------------------------------
  const int d    = tid & (HD - 1);
  const int half = tid >> 7;
  const float* vbase = cache_v + ((size_t)(kv * BATCH + b)) * WIN * HD;
  const float* vnew  = qkv + (size_t)b * QKVCOLS + (NHEADS + NKV) * HD + kv * HD;

  float o0 = 0.f, o1 = 0.f, o2 = 0.f, o3 = 0.f;
  const int s0 = half * (WIN / 2), s1 = s0 + (WIN / 2);
  for (int s = s0; s < s1; ++s) {
    const float vv = (s == cur) ? vnew[d] : vbase[(size_t)s * HD + d];
    o0 = fmaf(scores[0 * WIN + s], vv, o0);
    o1 = fmaf(scores[1 * WIN + s], vv, o1);
    o2 = fmaf(scores[2 * WIN + s], vv, o2);
    o3 = fmaf(scores[3 * WIN + s], vv, o3);
  }
  // reuse sums[] region as 4 x 2 x 128 combine buffer (after barrier)
  float* obuf = sums;
  obuf[(0 * 2 + half) * HD + d] = o0;
  obuf[(1 * 2 + half) * HD + d] = o1;
  obuf[(2 * 2 + half) * HD + d] = o2;
  obuf[(3 * 2 + half) * HD + d] = o3;
  __syncthreads();
  if (half == 0) {
#pragma unroll
    for (int g = 0; g < 4; ++g) {
      const float v = (obuf[(g * 2) * HD + d] + obuf[(g * 2 + 1) * HD + d]) * inv[g];
      attn_out[(size_t)b * DIMSZ + (size_t)(kv * 4 + g) * HD + d] = v;
    }
  }
}

// ---------------------------------------------------------------------------
extern "C" void kernel_launch(void* const* d_in, const int* in_sizes, int n_in,
                              void* d_out, int out_size, void* d_ws, size_t ws_size,
                              hipStream_t stream) {
  const float* x       = (const float*)d_in[0];  // 32 x 4096
  const float* wqkv    = (const float*)d_in[1];  // 4096 x 6144
  const float* wo      = (const float*)d_in[2];  // 4096 x 4096
  const float* rot     = (const float*)d_in[3];  // 128 x 128
  const float* cache_k = (const float*)d_in[4];  // 8 x 32 x 4096 x 128
  const float* cache_v = (const float*)d_in[5];
  const int*   spos    = (const int*)d_in[6];
  float* out = (float*)d_out;
  float* ws  = (float*)d_ws;

  float* qkv  = ws;                                   // 32*6144
  float* qrot = qkv + (size_t)BATCH * QKVCOLS;        // 32*4096
  float* krot = qrot + (size_t)BATCH * NHEADS * HD;   // 32*1024
  float* attn = krot + (size_t)BATCH * NKV * HD;      // 32*4096

  // 1) QKV projection: 6144 cols -> 192 chunks of 32 -> 192 blocks x 8 waves
  gemm32_wmma<<<QKVCOLS / 32, 256, 0, stream>>>(x, wqkv, qkv, DIMSZ, QKVCOLS);

  // 2) rotary on q (scaled) and k
  rotary_kernel<<<32, 128, 0, stream>>>(qkv, rot, qrot, krot);

  // 3) attention: one block per (kv_head, batch); 71 KB dynamic LDS
  const size_t smem = (size_t)(4 * WIN + 512 + 256 + 1024 + 8) * sizeof(float);
  attn_kernel<<<NKV * BATCH, 256, smem, stream>>>(qrot, krot, qkv, cache_k,
                                                  cache_v, spos, attn);

  // 4) output projection: 4096 cols -> 128 blocks x 8 waves
  gemm32_wmma<<<DIMSZ / 32, 256, 0, stream>>>(attn, wo, out, DIMSZ, DIMSZ);
}